// GCNEncoder_87557203296803
// MI455X (gfx1250) — compile-verified
//
#include <hip/hip_runtime.h>
#include <hip/hip_bf16.h>

// ---------------- problem constants (match reference) ----------------
#define NTOT  65536          // B*N total nodes
#define EDG   1048576        // total edges
#define KB1   820            // kept per graph after pool1
#define KB2   656            // kept per graph after pool2
#define NK1   (64*KB1)       // 52480
#define NK2   (64*KB2)       // 41984
#define BNEPS 1e-5f

typedef __attribute__((ext_vector_type(2))) float v2f;
typedef __attribute__((ext_vector_type(4))) float v4f;
typedef __attribute__((ext_vector_type(8))) float v8f;

__device__ __forceinline__ void atomAddF(float* p, float v) {
    __hip_atomic_fetch_add(p, v, __ATOMIC_RELAXED, __HIP_MEMORY_SCOPE_AGENT);
}

// ---------------- degree / normalization ----------------
__global__ __launch_bounds__(256) void k_deg_init(float* deg, int n) {
    int i = blockIdx.x * 256 + threadIdx.x;
    if (i < n) deg[i] = 1.0f;                       // self-loop weight 1
}

__global__ __launch_bounds__(256) void k_deg_scatter(const int* __restrict__ dst,
                                                     const float* __restrict__ ew,
                                                     float* deg, int e) {
    int i = blockIdx.x * 256 + threadIdx.x;
    if (i >= e) return;
    float w = ew[i];
    if (w != 0.0f) atomAddF(&deg[dst[i]], w);
}

__global__ __launch_bounds__(256) void k_deg_inv(const float* __restrict__ deg,
                                                 float* __restrict__ inv,
                                                 float* __restrict__ rinv, int n) {
    int i = blockIdx.x * 256 + threadIdx.x;
    if (i >= n) return;
    float d = deg[i];
    inv[i]  = 1.0f / d;
    rinv[i] = rsqrtf(d);
}

// ---------------- dense GEMM: Y[M x 64] = X[M x 64] * W[64 x 64] -------------
// V_WMMA_F32_16X16X4_F32, wave32. Each wave owns a 16-row strip and all 4
// column tiles (4 x v8f accumulators); K unrolled in steps of 4.
// W is staged in LDS pre-swizzled so each lane's B fragment (W[k][c],W[k+1][c])
// is one aligned 8-byte ds_load_b64:   Wl[(k>>1)*128 + c*2 + (k&1)]
// Also writes Oinit = Y * inv[row] + bias[col]  (self-loop term of GCNConv).
__global__ __launch_bounds__(256) void k_gemm64(const float* __restrict__ X,
                                                const float* __restrict__ W,
                                                float* __restrict__ Y,
                                                float* __restrict__ Oinit,
                                                const float* __restrict__ inv,
                                                const float* __restrict__ bias,
                                                int M) {
    __shared__ float Wl[64 * 64];
    int t = threadIdx.x;
    for (int i = t; i < 64 * 64; i += 256) {
        int k = i >> 6, c = i & 63;
        Wl[((k >> 1) * 64 + c) * 2 + (k & 1)] = W[i];
    }
    __syncthreads();

    int wave = t >> 5;
    int lane = t & 31;
    int row0 = (blockIdx.x * 8 + wave) * 16;
    if (row0 >= M) return;

    int mrow  = row0 + (lane & 15);
    int khalf = (lane >> 4) * 2;      // 0 or 2 : K sub-pair owned by this lane
    int col   = lane & 15;

    v8f c0 = {}, c1 = {}, c2 = {}, c3 = {};
    const float* xrow = X + (size_t)mrow * 64;
    #pragma unroll
    for (int k0 = 0; k0 < 64; k0 += 4) {
        v2f a = *(const v2f*)(xrow + k0 + khalf);          // 8B-aligned (k0+khalf even)
        int kp = (k0 + khalf) >> 1;                        // K-pair index 0..31
        const v2f* wp = (const v2f*)(Wl + kp * 128);
        v2f b0 = wp[col];
        v2f b1 = wp[col + 16];
        v2f b2 = wp[col + 32];
        v2f b3 = wp[col + 48];
        c0 = __builtin_amdgcn_wmma_f32_16x16x4_f32(false, a, false, b0, (short)0, c0, false, false);
        c1 = __builtin_amdgcn_wmma_f32_16x16x4_f32(false, a, false, b1, (short)0, c1, false, false);
        c2 = __builtin_amdgcn_wmma_f32_16x16x4_f32(false, a, false, b2, (short)0, c2, false, false);
        c3 = __builtin_amdgcn_wmma_f32_16x16x4_f32(false, a, false, b3, (short)0, c3, false, false);
    }

    float bi0 = bias[col];
    float bi1 = bias[col + 16];
    float bi2 = bias[col + 32];
    float bi3 = bias[col + 48];

    int mbase = row0 + (lane >> 4) * 8;           // C/D layout: lanes>=16 hold M+8
    #pragma unroll
    for (int m = 0; m < 8; ++m) {
        int r = mbase + m;
        float* yr = Y + (size_t)r * 64 + col;
        yr[0]  = c0[m];
        yr[16] = c1[m];
        yr[32] = c2[m];
        yr[48] = c3[m];
        float s = inv[r];
        float* orow = Oinit + (size_t)r * 64 + col;
        orow[0]  = c0[m] * s + bi0;
        orow[16] = c1[m] * s + bi1;
        orow[32] = c2[m] * s + bi2;
        orow[48] = c3[m] * s + bi3;
    }
}

// ---------------- edge feature scatter: Out[dst] += coef * H[src] ------------
__global__ __launch_bounds__(256) void k_feat_edges(const int* __restrict__ src,
                                                    const int* __restrict__ dst,
                                                    const float* __restrict__ ew,
                                                    const float* __restrict__ rinv,
                                                    const float* __restrict__ H,
                                                    float* __restrict__ Out, int e) {
    int eid = blockIdx.x * 4 + (threadIdx.x >> 6);
    int f   = threadIdx.x & 63;
    if (eid >= e) return;
    float w = ew[eid];
    if (w == 0.0f) return;                       // zeroed edges contribute nothing
    int s = src[eid], d = dst[eid];
    float coef = rinv[s] * w * rinv[d];
    atomAddF(&Out[(size_t)d * 64 + f], coef * H[(size_t)s * 64 + f]);
}

// ---------------- batchnorm (2-pass) + PReLU ----------------
__global__ __launch_bounds__(256) void k_zero(float* p, int n) {
    int i = blockIdx.x * 256 + threadIdx.x;
    if (i < n) p[i] = 0.0f;
}

__global__ __launch_bounds__(256) void k_bn_reduce(const float* __restrict__ X,
                                                   float* __restrict__ acc, int M) {
    __shared__ float ls[256], lq[256];
    int f   = threadIdx.x & 63;
    int sub = threadIdx.x >> 6;
    int r0  = blockIdx.x * 256;
    int re  = r0 + 256; if (re > M) re = M;
    float s = 0.0f, q = 0.0f;
    for (int r = r0 + sub; r < re; r += 4) {
        float v = X[(size_t)r * 64 + f];
        s += v; q += v * v;
    }
    ls[threadIdx.x] = s; lq[threadIdx.x] = q;
    __syncthreads();
    if (sub == 0) {
        s = ls[f] + ls[f + 64] + ls[f + 128] + ls[f + 192];
        q = lq[f] + lq[f + 64] + lq[f + 128] + lq[f + 192];
        atomAddF(&acc[f], s);
        atomAddF(&acc[64 + f], q);
    }
}

__global__ void k_bn_final(float* acc, const float* __restrict__ g,
                           const float* __restrict__ be, float fM) {
    int f = threadIdx.x;                         // 64 threads
    float mean = acc[f] / fM;
    float var  = acc[64 + f] / fM - mean * mean; // biased variance (jnp.var)
    float sc   = g[f] * rsqrtf(var + BNEPS);
    acc[128 + f] = sc;
    acc[192 + f] = be[f] - mean * sc;
}

__global__ __launch_bounds__(256) void k_bn_prelu(float* __restrict__ X,
                                                  const float* __restrict__ acc,
                                                  const float* __restrict__ alpha, int n) {
    int i = blockIdx.x * 256 + threadIdx.x;
    if (i >= n) return;
    int f = i & 63;
    float y = acc[128 + f] * X[i] + acc[192 + f];
    X[i] = (y >= 0.0f) ? y : alpha[0] * y;
}

// ---------------- score GEMV (d -> 1 GCNConv, node part) ----------------
__global__ __launch_bounds__(256) void k_gemv64(const float* __restrict__ X,
                                                const float* __restrict__ Wsc,
                                                const float* __restrict__ inv,
                                                const float* __restrict__ bsc,
                                                float* __restrict__ slin,
                                                float* __restrict__ score, int M) {
    int r = blockIdx.x * 256 + threadIdx.x;
    if (r >= M) return;
    const v4f* xr = (const v4f*)(X + (size_t)r * 64);
    const v4f* wr = (const v4f*)Wsc;
    float acc = 0.0f;
    #pragma unroll
    for (int k = 0; k < 16; ++k) {
        v4f xv = xr[k], wv = wr[k];
        acc += xv.x * wv.x + xv.y * wv.y + xv.z * wv.z + xv.w * wv.w;
    }
    slin[r]  = acc;
    score[r] = acc * inv[r] + bsc[0];
}

__global__ __launch_bounds__(256) void k_score_edges(const int* __restrict__ src,
                                                     const int* __restrict__ dst,
                                                     const float* __restrict__ ew,
                                                     const float* __restrict__ rinv,
                                                     const float* __restrict__ slin,
                                                     float* __restrict__ score, int e) {
    int i = blockIdx.x * 256 + threadIdx.x;
    if (i >= e) return;
    float w = ew[i];
    if (w == 0.0f) return;
    int s = src[i], d = dst[i];
    atomAddF(&score[d], rinv[s] * w * rinv[d] * slin[s]);
}

// ---------------- per-graph top-k via bitonic sort (descending, idx tiebreak)
__global__ __launch_bounds__(1024) void k_topk(const float* __restrict__ score,
                                               float* __restrict__ gate,
                                               int* __restrict__ perm,
                                               int* __restrict__ mapping,
                                               int n_per, int k) {
    __shared__ float ss[1024];
    __shared__ int   si[1024];
    int g = blockIdx.x;
    int t = threadIdx.x;
    if (t < n_per) { ss[t] = score[g * n_per + t]; si[t] = t; }
    else           { ss[t] = -__builtin_inff();    si[t] = 0x7fffffff; }
    __syncthreads();
    for (int kk = 2; kk <= 1024; kk <<= 1) {
        for (int j = kk >> 1; j > 0; j >>= 1) {
            int x = t ^ j;
            if (x > t) {
                float a = ss[t], b = ss[x];
                int   ia = si[t], ib = si[x];
                bool aFirst = (a > b) || (a == b && ia < ib);   // descending order
                bool desc   = ((t & kk) == 0);
                if (desc ? !aFirst : aFirst) {
                    ss[t] = b; ss[x] = a; si[t] = ib; si[x] = ia;
                }
            }
            __syncthreads();
        }
    }
    if (t < n_per) {
        int oldIdx = si[t];                      // every real node appears once
        if (t < k) {
            perm[g * k + t]              = g * n_per + oldIdx;
            gate[g * k + t]              = tanhf(ss[t]);
            mapping[g * n_per + oldIdx]  = g * k + t;
        } else {
            mapping[g * n_per + oldIdx]  = -1;
        }
    }
}

__global__ __launch_bounds__(256) void k_gather_mul(const float* __restrict__ X,
                                                    const int* __restrict__ perm,
                                                    const float* __restrict__ gate,
                                                    float* __restrict__ Y, int nk) {
    int i = blockIdx.x * 256 + threadIdx.x;
    if (i >= nk * 64) return;
    int r = i >> 6, f = i & 63;
    Y[i] = X[(size_t)perm[r] * 64 + f] * gate[r];
}

__global__ __launch_bounds__(256) void k_remap_edges(const int* __restrict__ src,
                                                     const int* __restrict__ dst,
                                                     const float* __restrict__ ew,
                                                     const int* __restrict__ mapping,
                                                     int* __restrict__ nsrc,
                                                     int* __restrict__ ndst,
                                                     float* __restrict__ new_ew, int e) {
    int i = blockIdx.x * 256 + threadIdx.x;
    if (i >= e) return;
    int ns = mapping[src[i]], nd = mapping[dst[i]];
    bool valid = (ns >= 0) && (nd >= 0);
    new_ew[i] = valid ? ew[i] : 0.0f;
    nsrc[i]   = valid ? ns : 0;
    ndst[i]   = valid ? nd : 0;
}

// ---------------- global mean pool over kept nodes ----------------
__global__ void k_mean_pool(const float* __restrict__ H, float* __restrict__ Z, int k) {
    int g = blockIdx.x;
    int f = threadIdx.x;                         // 64 threads (coalesced per row)
    float s = 0.0f;
    for (int j = 0; j < k; ++j) s += H[((size_t)g * k + j) * 64 + f];
    Z[g * 64 + f] = s / (float)k;
}

// ---------------- launch ----------------
extern "C" void kernel_launch(void* const* d_in, const int* in_sizes, int n_in,
                              void* d_out, int out_size, void* d_ws, size_t ws_size,
                              hipStream_t stream) {
    const float* x   = (const float*)d_in[0];
    const int*   ei  = (const int*)  d_in[1];
    const float* ew  = (const float*)d_in[2];
    const float* W1  = (const float*)d_in[4];
    const float* b1  = (const float*)d_in[5];
    const float* g1  = (const float*)d_in[6];
    const float* be1 = (const float*)d_in[7];
    const float* a1  = (const float*)d_in[8];
    const float* Ws1 = (const float*)d_in[9];
    const float* bs1 = (const float*)d_in[10];
    const float* W2  = (const float*)d_in[11];
    const float* b2  = (const float*)d_in[12];
    const float* g2  = (const float*)d_in[13];
    const float* be2 = (const float*)d_in[14];
    const float* a2  = (const float*)d_in[15];
    const float* Ws2 = (const float*)d_in[16];
    const float* bs2 = (const float*)d_in[17];
    const int* srcI = ei;
    const int* dstI = ei + EDG;

    // workspace carve-up (256B aligned)
    char* p = (char*)d_ws;
    auto carve = [&](size_t bytes) -> void* {
        void* r = (void*)p;
        p += (bytes + 255) & ~(size_t)255;
        return r;
    };
    float* Hlin  = (float*)carve(sizeof(float) * (size_t)NTOT * 64);
    float* Hout  = (float*)carve(sizeof(float) * (size_t)NTOT * 64);
    float* deg   = (float*)carve(sizeof(float) * NTOT);
    float* inv   = (float*)carve(sizeof(float) * NTOT);
    float* rinv  = (float*)carve(sizeof(float) * NTOT);
    float* slin  = (float*)carve(sizeof(float) * NTOT);
    float* score = (float*)carve(sizeof(float) * NTOT);
    float* gate  = (float*)carve(sizeof(float) * NTOT);
    int*   perm  = (int*)  carve(sizeof(int)   * NTOT);
    int*   remap = (int*)  carve(sizeof(int)   * NTOT);
    int*   src2  = (int*)  carve(sizeof(int)   * EDG);
    int*   dst2  = (int*)  carve(sizeof(int)   * EDG);
    float* ew2   = (float*)carve(sizeof(float) * EDG);
    float* Xn1   = (float*)carve(sizeof(float) * (size_t)NK1 * 64);
    float* bnb   = (float*)carve(sizeof(float) * 256);
    float* hfin  = (float*)d_out;
    float* zout  = hfin + (size_t)NK2 * 64;

    // ---------------- layer 1 (nodes NTOT, original edges) ----------------
    k_deg_init   <<<NTOT / 256, 256, 0, stream>>>(deg, NTOT);
    k_deg_scatter<<<EDG / 256, 256, 0, stream>>>(dstI, ew, deg, EDG);
    k_deg_inv    <<<NTOT / 256, 256, 0, stream>>>(deg, inv, rinv, NTOT);
    k_gemm64     <<<NTOT / 128, 256, 0, stream>>>(x, W1, Hlin, Hout, inv, b1, NTOT);
    k_feat_edges <<<EDG / 4, 256, 0, stream>>>(srcI, dstI, ew, rinv, Hlin, Hout, EDG);
    k_zero       <<<1, 256, 0, stream>>>(bnb, 256);
    k_bn_reduce  <<<NTOT / 256, 256, 0, stream>>>(Hout, bnb, NTOT);
    k_bn_final   <<<1, 64, 0, stream>>>(bnb, g1, be1, (float)NTOT);
    k_bn_prelu   <<<NTOT * 64 / 256, 256, 0, stream>>>(Hout, bnb, a1, NTOT * 64);

    // ---------------- pool 1 (keep 820/1024 per graph) ----------------
    k_gemv64     <<<NTOT / 256, 256, 0, stream>>>(Hout, Ws1, inv, bs1, slin, score, NTOT);
    k_score_edges<<<EDG / 256, 256, 0, stream>>>(srcI, dstI, ew, rinv, slin, score, EDG);
    k_topk       <<<64, 1024, 0, stream>>>(score, gate, perm, remap, 1024, KB1);
    k_gather_mul <<<(NK1 * 64) / 256, 256, 0, stream>>>(Hout, perm, gate, Xn1, NK1);
    k_remap_edges<<<EDG / 256, 256, 0, stream>>>(srcI, dstI, ew, remap, src2, dst2, ew2, EDG);

    // ---------------- layer 2 (nodes NK1, filtered edges) ----------------
    k_deg_init   <<<NK1 / 256, 256, 0, stream>>>(deg, NK1);
    k_deg_scatter<<<EDG / 256, 256, 0, stream>>>(dst2, ew2, deg, EDG);
    k_deg_inv    <<<NK1 / 256, 256, 0, stream>>>(deg, inv, rinv, NK1);
    k_gemm64     <<<NK1 / 128, 256, 0, stream>>>(Xn1, W2, Hlin, Hout, inv, b2, NK1);
    k_feat_edges <<<EDG / 4, 256, 0, stream>>>(src2, dst2, ew2, rinv, Hlin, Hout, EDG);
    k_zero       <<<1, 256, 0, stream>>>(bnb, 256);
    k_bn_reduce  <<<NK1 / 256, 256, 0, stream>>>(Hout, bnb, NK1);
    k_bn_final   <<<1, 64, 0, stream>>>(bnb, g2, be2, (float)NK1);
    k_bn_prelu   <<<NK1 * 64 / 256, 256, 0, stream>>>(Hout, bnb, a2, NK1 * 64);

    // ---------------- pool 2 (keep 656/820 per graph) ----------------
    k_gemv64     <<<NK1 / 256, 256, 0, stream>>>(Hout, Ws2, inv, bs2, slin, score, NK1);
    k_score_edges<<<EDG / 256, 256, 0, stream>>>(src2, dst2, ew2, rinv, slin, score, EDG);
    k_topk       <<<64, 1024, 0, stream>>>(score, gate, perm, remap, KB1, KB2);
    k_gather_mul <<<(NK2 * 64) / 256, 256, 0, stream>>>(Hout, perm, gate, hfin, NK2);

    // ---------------- readout ----------------
    k_mean_pool  <<<64, 64, 0, stream>>>(hfin, zout, KB2);

    (void)in_sizes; (void)n_in; (void)out_size; (void)ws_size;
}